// GraphConv_43207370998362
// MI455X (gfx1250) — compile-verified
//
#include <hip/hip_runtime.h>
#include <hip/hip_bf16.h>
#include <cstdint>

#define CHUNK 256
#define WPB   8    // waves per 256-thread block (wave32)

// ---- CDNA5 async global->LDS copy (gfx1250), GVS addressing:
//      mem_addr = SGPR64 + VGPR32 ; lds_addr = wave LDS base + VGPR(vdst)
//      Edge lists are touched exactly once -> non-temporal so they don't
//      evict the L2-resident ego/out working set (205MB vs 192MB L2).
__device__ __forceinline__ void async_cp_b32_nt(void* lds_dst, const void* gbase,
                                                unsigned voff_bytes) {
    unsigned lds = (unsigned)(uintptr_t)lds_dst;   // low 32 bits of generic LDS addr
    asm volatile("global_load_async_to_lds_b32 %0, %1, %2 th:TH_LOAD_NT"
                 :: "v"(lds), "v"(voff_bytes), "s"(gbase)
                 : "memory");
}
__device__ __forceinline__ void wait_async_le0() {
    asm volatile("s_wait_asynccnt 0x0" ::: "memory");
}
__device__ __forceinline__ void wait_async_le3() {
    asm volatile("s_wait_asynccnt 0x3" ::: "memory");
}

__device__ __forceinline__ void atomic_add_f32(float* p, float v) {
    // relaxed, device scope, result unused -> non-returning global_atomic_add_f32
    __hip_atomic_fetch_add(p, v, __ATOMIC_RELAXED, __HIP_MEMORY_SCOPE_AGENT);
}

// ---------------- SpMM: wave-per-edge, async double-buffered edge lists ------
// (defined FIRST so the disasm snippet shows this kernel's vmem instructions)
__global__ __launch_bounds__(256)
void spmm_coo_kernel(const float* __restrict__ ego,     // [nNodes,128]
                     const float* __restrict__ vals,    // [nnz]
                     const int*   __restrict__ rows,    // [nnz]
                     const int*   __restrict__ cols,    // [nnz]
                     float*       __restrict__ out,     // [nNodes,128]
                     long long nnz, long long numChunks)
{
    __shared__ int   sRow[2][CHUNK];
    __shared__ int   sCol[2][CHUNK];
    __shared__ float sVal[2][CHUNK];

    const int t    = threadIdx.x;
    const int lane = t & 31;
    const int wav  = t >> 5;

    // Thread t stages edge (chunkBase + t); wave w consumes edges
    // [w*32, w*32+32) of the chunk == exactly its own lanes' loads, so the
    // only synchronization needed is this wave's own ASYNCcnt (no barriers).
    auto issue = [&](long long c, int b) {
        unsigned off = (unsigned)((c * CHUNK + t) * 4u);
        async_cp_b32_nt(&sRow[b][t], rows, off);
        async_cp_b32_nt(&sCol[b][t], cols, off);
        async_cp_b32_nt(&sVal[b][t], vals, off);
    };

    long long c  = blockIdx.x;
    int buf = 0;
    if (c < numChunks) issue(c, 0);

    for (; c < numChunks; c += gridDim.x) {
        long long cn = c + gridDim.x;
        const bool more = (cn < numChunks);
        if (more) { issue(cn, buf ^ 1); wait_async_le3(); }  // newest 3 = next chunk
        else      { wait_async_le0(); }

        const int ebase = wav * 32;
        #pragma unroll 8
        for (int j = 0; j < 32; ++j) {
            const int e = ebase + j;
            if (j + 8 < 32) {   // deepen gather pipeline: global_prefetch_b8
                const int pcol = sCol[buf][ebase + j + 8];
                __builtin_prefetch(ego + (size_t)pcol * 128 + lane * 4, 0, 3);
            }
            const int   row = sRow[buf][e];
            const int   col = sCol[buf][e];
            const float val = sVal[buf][e];

            const float4 v = *reinterpret_cast<const float4*>(
                ego + (size_t)col * 128 + (size_t)lane * 4);   // global_load_b128
            float* o = out + (size_t)row * 128 + (size_t)lane * 4;
            atomic_add_f32(o + 0, v.x * val);
            atomic_add_f32(o + 1, v.y * val);
            atomic_add_f32(o + 2, v.z * val);
            atomic_add_f32(o + 3, v.w * val);
        }
        buf ^= 1;
    }

    // Generic tail (nnz not multiple of CHUNK): direct global loads, wave/edge.
    const long long tailStart = numChunks * CHUNK;
    if (tailStart < nnz) {
        const long long gw = (long long)blockIdx.x * WPB + wav;
        const long long nw = (long long)gridDim.x * WPB;
        for (long long e = tailStart + gw; e < nnz; e += nw) {
            const int   row = rows[e];
            const int   col = cols[e];
            const float val = vals[e];
            const float4 v = *reinterpret_cast<const float4*>(
                ego + (size_t)col * 128 + (size_t)lane * 4);
            float* o = out + (size_t)row * 128 + (size_t)lane * 4;
            atomic_add_f32(o + 0, v.x * val);
            atomic_add_f32(o + 1, v.y * val);
            atomic_add_f32(o + 2, v.z * val);
            atomic_add_f32(o + 3, v.w * val);
        }
    }
}

// ---------------- zero-init of the output (harness poisons d_out) -------------
__global__ __launch_bounds__(256) void zero_f32_kernel(float4* __restrict__ out,
                                                       long long n4) {
    long long i = (long long)blockIdx.x * blockDim.x + threadIdx.x;
    long long stride = (long long)gridDim.x * blockDim.x;
    float4 z; z.x = 0.f; z.y = 0.f; z.z = 0.f; z.w = 0.f;
    for (; i < n4; i += stride) out[i] = z;
}

extern "C" void kernel_launch(void* const* d_in, const int* in_sizes, int n_in,
                              void* d_out, int out_size, void* d_ws, size_t ws_size,
                              hipStream_t stream) {
    const float* ego  = (const float*)d_in[0];   // [nNodes*128]
    const float* vals = (const float*)d_in[1];   // [nnz]
    const int*   rows = (const int*)d_in[2];     // [nnz]
    const int*   cols = (const int*)d_in[3];     // [nnz]
    float*       out  = (float*)d_out;           // [nNodes*128]

    const long long nnz   = (long long)in_sizes[1];
    const long long outN  = (long long)out_size;          // nNodes*128
    const long long n4    = outN / 4;

    // 1) zero the accumulator
    int zblocks = (int)((n4 + 255) / 256);
    if (zblocks > 4096) zblocks = 4096;
    if (zblocks < 1)    zblocks = 1;
    zero_f32_kernel<<<zblocks, 256, 0, stream>>>((float4*)out, n4);

    // 2) scatter-add SpMM
    const long long numChunks = nnz / CHUNK;
    int blocks = 2048;
    if (numChunks > 0 && numChunks < blocks) blocks = (int)numChunks;
    if (blocks < 1) blocks = 1;
    spmm_coo_kernel<<<blocks, 256, 0, stream>>>(ego, vals, rows, cols, out,
                                                nnz, numChunks);
}